// RoIAlignRotatedV2_24816321036756
// MI455X (gfx1250) — compile-verified
//
#include <hip/hip_runtime.h>
#include <hip/hip_bf16.h>
#include <stdint.h>

// RoIAlignRotated for MI455X (gfx1250, wave32).
// One block per ROI, 256 threads (8 waves), thread == channel.
// Phase 1: 196 threads build per-(bin,sample) descriptors in LDS
//          (pair-load offsets + weights with valid/mean/clamp folded in).
// Phase 2: each thread gathers with global_load_b64 pair loads (x-corners are
//          adjacent), accumulates one bin at a time, writes its 49 outputs
//          into the LDS tile in final [c][bin] layout.
// Phase 3: each WAVE issues its own Tensor-Data-Mover tensor_store_from_lds
//          for its contiguous 32-channel slice (6272 B) as soon as it finishes
//          -- no trailing barrier, store DMA overlaps other waves' compute.

#define N_CH     256
#define H_FEAT   256
#define W_FEAT   256
#define PH_      7
#define PW_      7
#define NBINS    49          // PH*PW
#define NSAMP    196         // NBINS * 2 * 2
#define OUT_PER_ROI 12544    // N_CH * NBINS
#define PLANE    65536       // H*W
#define WAVE_ELEMS 1568      // 32 channels * 49 bins per wave slice (0x620)

typedef unsigned int u32x4 __attribute__((ext_vector_type(4)));
typedef int          i32x8 __attribute__((ext_vector_type(8)));
typedef int          i32x4 __attribute__((ext_vector_type(4)));

__global__ __launch_bounds__(256)
void roi_align_rotated_kernel(const float* __restrict__ feat,
                              const float* __restrict__ rois,
                              float* __restrict__ out)
{
    __shared__ int2   sh_off[NSAMP];   // {off_top, off_bot} element offsets in plane
    __shared__ float4 sh_w[NSAMP];     // weights for {top.x, top.y, bot.x, bot.y}
    __shared__ int    sh_b;            // batch index
    __shared__ float  lds_out[OUT_PER_ROI];  // 49KB output tile, [c][bin] = final layout

    const int roi = blockIdx.x;
    const int tid = threadIdx.x;

    // ---------------- Phase 1: descriptor build (196 threads) ----------------
    if (tid < NSAMP) {
        const float* r = rois + roi * 6;
        const float r0 = r[0], r1 = r[1], r2 = r[2], r3 = r[3], r4 = r[4], r5 = r[5];
        if (tid == 0) sh_b = (int)r0;

        const float cx = r1 * 0.25f - 0.5f;   // SPATIAL_SCALE=0.25, ALIGNED offset
        const float cy = r2 * 0.25f - 0.5f;
        const float rw = r3 * 0.25f;
        const float rh = r4 * 0.25f;
        const float th = r5;                  // CLOCKWISE = False

        const float bin_h = rh / (float)PH_;
        const float bin_w = rw / (float)PW_;

        const int bin  = tid >> 2;            // 0..48
        const int s    = tid & 3;             // sample within bin
        const int ph   = bin / PW_;
        const int pw   = bin % PW_;
        const int sy_i = s >> 1;
        const int sx_i = s & 1;

        // (py + (sy+0.5)/gh) with gh=2  ->  py + 0.5*sy + 0.25
        const float yy = -0.5f * rh + ((float)ph + 0.5f * (float)sy_i + 0.25f) * bin_h;
        const float xx = -0.5f * rw + ((float)pw + 0.5f * (float)sx_i + 0.25f) * bin_w;

        const float cs = cosf(th), sn = sinf(th);
        float y = yy * cs - xx * sn + cy;
        float x = yy * sn + xx * cs + cx;

        const bool valid = (y > -1.0f) && (y < (float)H_FEAT) &&
                           (x > -1.0f) && (x < (float)W_FEAT);
        y = fmaxf(y, 0.0f);
        x = fmaxf(x, 0.0f);

        int yl = (int)y;
        int xl = (int)x;
        if (yl >= H_FEAT - 1) { yl = H_FEAT - 1; y = (float)yl; }
        if (xl >= W_FEAT - 1) { xl = W_FEAT - 1; x = (float)xl; }
        const int yh = min(yl + 1, H_FEAT - 1);

        const float ly = y - (float)yl, lx = x - (float)xl;
        const float hy = 1.0f - ly,     hx = 1.0f - lx;

        // Fold x-clamp into a pair load at x0 = min(xl, W-2):
        //   normal: pair = {v(xl), v(xl+1)}, weights (hx, lx)
        //   clamp : pair = {v(W-2), v(W-1)}, both x-terms hit pair[1],
        //           weight (0, hx+lx) (lx==0 so == hx)
        const int  x0   = min(xl, W_FEAT - 2);
        const bool selx = (xl > W_FEAT - 2);
        const float wxA = selx ? 0.0f : hx;
        const float wxB = selx ? (hx + lx) : lx;

        const float sc = valid ? 0.25f : 0.0f;   // valid mask * 1/(gh*gw)

        float4 w;
        w.x = sc * hy * wxA;   // top row, pair[0]
        w.y = sc * hy * wxB;   // top row, pair[1]
        w.z = sc * ly * wxA;   // bottom row, pair[0]
        w.w = sc * ly * wxB;   // bottom row, pair[1]
        sh_w[tid]   = w;
        sh_off[tid] = make_int2(yl * W_FEAT + x0, yh * W_FEAT + x0);
    }
    __syncthreads();

    // ---------------- Phase 2: gather + accumulate (thread == channel) --------
    const int c = tid;
    const float* plane = feat + ((size_t)sh_b * N_CH + c) * PLANE;

    for (int bin = 0; bin < NBINS; ++bin) {
        // Prefetch next bin's first sample rows into cache (global_prefetch_b8).
        if (bin + 1 < NBINS) {
            int2 o = sh_off[(bin + 1) * 4];
            __builtin_prefetch(plane + o.x, 0, 3);
            __builtin_prefetch(plane + o.y, 0, 3);
        }
        float acc = 0.0f;
#pragma unroll
        for (int s = 0; s < 4; ++s) {
            const int d = bin * 4 + s;
            const int2   o = sh_off[d];   // wave-uniform LDS broadcast
            const float4 w = sh_w[d];
            const float2 t = *(const float2*)(plane + o.x);  // global_load_b64
            const float2 b = *(const float2*)(plane + o.y);  // global_load_b64
            acc = fmaf(w.x, t.x, acc);
            acc = fmaf(w.y, t.y, acc);
            acc = fmaf(w.z, b.x, acc);
            acc = fmaf(w.w, b.y, acc);
        }
        lds_out[c * NBINS + bin] = acc;   // final [roi][c][ph][pw] layout in LDS
    }

    // ---------------- Phase 3: per-wave TDM store LDS -> global ---------------
    // Wave w owns channels [w*32, w*32+32) -> contiguous WAVE_ELEMS floats of
    // lds_out and of the output. Each wave DMAs its slice as soon as it is
    // done; no barrier needed, stores overlap remaining compute of other waves.
#ifdef __HIP_DEVICE_COMPILE__
    {
        // Drain this wave's LDS stores before the TDM engine reads them
        // (tensor ops are unordered with DS ops).
        asm volatile("s_wait_dscnt 0" ::: "memory");

        const int wave = tid >> 5;   // wave-uniform
        const uint64_t gaddr =
            (uint64_t)(uintptr_t)(out + (size_t)roi * OUT_PER_ROI + wave * WAVE_ELEMS);
        const uint32_t lds_off =
            (uint32_t)(uintptr_t)(void*)(lds_out + wave * WAVE_ELEMS);

        // D# group 0: count=1, lds_addr, global_addr[56:0], type=2 (ISA §8.3)
        u32x4 g0 = { 1u,
                     lds_off,
                     (uint32_t)gaddr,
                     (uint32_t)((gaddr >> 32) & 0x01FFFFFFu) | (2u << 30) };

        // D# group 1 (ISA §8.4): data_size=2 (4B elements);
        // tensor_dim0 = tile_dim0 = strides = 1568 (0x620); tensor/tile_dim1=1.
        i32x8 g1 = { 0x00020000,    // wg_mask=0, data_size=4B
                     0x06200000,    // tensor_dim0[15:0]  << 16
                     0x00010000,    // tensor_dim0[31:16]=0 | tensor_dim1[15:0]=1 << 16
                     0x06200000,    // tensor_dim1[31:16]=0 | tile_dim0=1568 << 16
                     0x00000001,    // tile_dim1=1, tile_dim2=0
                     WAVE_ELEMS,    // tensor_dim0_stride[31:0]
                     0x06200000,    // stride0[47:32]=0 | tensor_dim1_stride[15:0] << 16
                     0 };           // tensor_dim1_stride[47:16]
        i32x4 gz = { 0, 0, 0, 0 };
#if __clang_major__ >= 23
        i32x8 gz8 = { 0, 0, 0, 0, 0, 0, 0, 0 };
        __builtin_amdgcn_tensor_store_from_lds(g0, g1, gz, gz, gz8, 0);
#else
        __builtin_amdgcn_tensor_store_from_lds(g0, g1, gz, gz, 0);
#endif
        __builtin_amdgcn_s_wait_tensorcnt(0);
    }
#endif
}

extern "C" void kernel_launch(void* const* d_in, const int* in_sizes, int n_in,
                              void* d_out, int out_size, void* d_ws, size_t ws_size,
                              hipStream_t stream) {
    const float* feat = (const float*)d_in[0];
    const float* rois = (const float*)d_in[1];
    float* out = (float*)d_out;
    const int n_rois = in_sizes[1] / 6;     // 1024
    roi_align_rotated_kernel<<<n_rois, 256, 0, stream>>>(feat, rois, out);
}